// GNNLayer_79207786873559
// MI455X (gfx1250) — compile-verified
//
#include <hip/hip_runtime.h>

typedef float v2f __attribute__((ext_vector_type(2)));
typedef float v8f __attribute__((ext_vector_type(8)));

#define DFEAT 128

// ---------------------------------------------------------------------------
// Kernel 1: P = node_emb @ W^T  (fp32 WMMA, 16x16x4_f32)
//   grid.x  = M/16 (one M-tile per block)
//   block   = 256 threads = 8 waves; wave w handles N-tile n0 = 16*w
//   Per wave: K-loop 0..127 step 4 -> 32 x v_wmma_f32_16x16x4_f32
// A-layout (16x4 f32): lanes 0-15 -> K=0,1 ; lanes 16-31 -> K=2,3 (per lane row M)
// B-layout (4x16 f32): symmetric, lane = column N
// C/D (16x16 f32, 8 VGPRs): VGPR v: lanes 0-15 M=v, lanes 16-31 M=v+8, N=lane%16
// ---------------------------------------------------------------------------
__global__ void __launch_bounds__(256)
gemm_emb_wt(const float* __restrict__ emb,
            const float* __restrict__ W,
            float* __restrict__ P) {
  const int wave = threadIdx.x >> 5;
  const int lane = threadIdx.x & 31;
  const int half = lane >> 4;       // 0: lanes 0-15, 1: lanes 16-31
  const int lrow = lane & 15;
  const int m0 = blockIdx.x * 16;
  const int n0 = wave * 16;

  // P[m][n] = sum_k emb[m][k] * W[n][k]   (B[k][n] = W[n][k], row-major W)
  const float* __restrict__ arow = emb + (size_t)(m0 + lrow) * DFEAT;
  const float* __restrict__ brow = W   + (size_t)(n0 + lrow) * DFEAT;

  v8f c = {};
#pragma unroll
  for (int k0 = 0; k0 < DFEAT; k0 += 4) {
    const int k = k0 + 2 * half;                 // 8B-aligned (k0 mult of 4)
    v2f a = *(const v2f*)(arow + k);
    v2f b = *(const v2f*)(brow + k);
    // (neg_a, A, neg_b, B, c_mod, C, reuse_a, reuse_b)
    c = __builtin_amdgcn_wmma_f32_16x16x4_f32(false, a, false, b,
                                              (short)0, c, false, false);
  }

#pragma unroll
  for (int v = 0; v < 8; ++v) {
    P[(size_t)(m0 + v + 8 * half) * DFEAT + (n0 + lrow)] = c[v];
  }
}

// ---------------------------------------------------------------------------
// Kernel 2: zero d_out (float4 stores)
// ---------------------------------------------------------------------------
__global__ void __launch_bounds__(256)
zero_f4(float4* __restrict__ p, long n4) {
  long i = (long)blockIdx.x * blockDim.x + threadIdx.x;
  if (i < n4) p[i] = make_float4(0.f, 0.f, 0.f, 0.f);
}

// ---------------------------------------------------------------------------
// Kernel 3: out[dst[e]] += edge_weight[e] * P[src[e]]
//   one wave32 per edge; each lane owns 4 consecutive floats (float4 load,
//   4 native f32 atomic adds -> global_atomic_add_f32, no return)
// ---------------------------------------------------------------------------
__global__ void __launch_bounds__(256)
scatter_edges(const float* __restrict__ P,
              const float* __restrict__ ew,
              const int*   __restrict__ src,
              const int*   __restrict__ dst,
              float*       __restrict__ out,
              int E) {
  const int wid  = (int)(((long)blockIdx.x * blockDim.x + threadIdx.x) >> 5);
  if (wid >= E) return;
  const int lane = threadIdx.x & 31;

  const int   s = src[wid];
  const int   d = dst[wid];
  const float w = ew[wid];

  const float4 v = *(const float4*)(P + (size_t)s * DFEAT + lane * 4);
  float* __restrict__ o = out + (size_t)d * DFEAT + lane * 4;

#if defined(__AMDGCN__)
  unsafeAtomicAdd(o + 0, v.x * w);
  unsafeAtomicAdd(o + 1, v.y * w);
  unsafeAtomicAdd(o + 2, v.z * w);
  unsafeAtomicAdd(o + 3, v.w * w);
#else
  atomicAdd(o + 0, v.x * w);
  atomicAdd(o + 1, v.y * w);
  atomicAdd(o + 2, v.z * w);
  atomicAdd(o + 3, v.w * w);
#endif
}

extern "C" void kernel_launch(void* const* d_in, const int* in_sizes, int n_in,
                              void* d_out, int out_size, void* d_ws, size_t ws_size,
                              hipStream_t stream) {
  const float* node_emb = (const float*)d_in[0];   // [N, 128] f32
  const float* edge_w   = (const float*)d_in[1];   // [E]      f32
  const int*   src      = (const int*)  d_in[2];   // [E]      i32
  const int*   dst      = (const int*)  d_in[3];   // [E]      i32
  const float* W        = (const float*)d_in[4];   // [128,128] f32

  const int N = in_sizes[0] / DFEAT;               // 100000
  const int E = in_sizes[1];                       // 1600000

  float* P   = (float*)d_ws;                       // [N, 128] scratch (51.2 MB)
  float* out = (float*)d_out;                      // [N, 128]

  // 1) P = node_emb @ W^T  (WMMA f32)
  {
    dim3 grid(N / 16);          // 6250 blocks; N is a multiple of 16
    dim3 block(256);            // 8 waves = 8 N-tiles of 16
    gemm_emb_wt<<<grid, block, 0, stream>>>(node_emb, W, P);
  }

  // 2) out = 0
  {
    const long n4 = (long)out_size / 4;
    dim3 grid((unsigned)((n4 + 255) / 256));
    zero_f4<<<grid, 256, 0, stream>>>((float4*)out, n4);
  }

  // 3) out[dst] += w * P[src]   (one wave per edge)
  {
    const long threads = (long)E * 32;
    dim3 grid((unsigned)((threads + 255) / 256));
    scatter_edges<<<grid, 256, 0, stream>>>(P, edge_w, src, dst, out, E);
  }
}